// GraphConvolution_36129264894614
// MI455X (gfx1250) — compile-verified
//
#include <hip/hip_runtime.h>
#include <hip/hip_bf16.h>

typedef __attribute__((ext_vector_type(2))) float v2f;
typedef __attribute__((ext_vector_type(8))) float v8f;

#define D_IN 256
#define D_OUT 128

// ---------------------------------------------------------------------------
// Phase 0: zero the output accumulator (harness poisons d_out with 0xAA).
// ---------------------------------------------------------------------------
__global__ void gc_zero_kernel(float4* __restrict__ out, int n4) {
    int i = blockIdx.x * blockDim.x + threadIdx.x;
    if (i < n4) out[i] = make_float4(0.f, 0.f, 0.f, 0.f);
}

// ---------------------------------------------------------------------------
// Phase 1: h = x @ W via V_WMMA_F32_16X16X4_F32 (fp32 matrix pipe, wave32).
// Block = 256 threads = 8 waves. Block owns a 16-row M tile; wave w owns the
// 16-col N tile [16w, 16w+16). K-loop: 256/4 = 64 WMMA steps.
//
// Fragment layouts (ISA 7.12.2, 32-bit, wave32):
//   A 16x4 : lane<16 -> M=lane,   K = {0,1}; lane>=16 -> M=lane-16, K = {2,3}
//            => per lane one contiguous 2-float (8B) load from x row.
//   B 4x16 : vreg g, half h -> K = g + 2h, N = n0 + (lane&15)
//   C 16x16: vreg g, half h -> M = g + 8h, N = n0 + (lane&15)
// ---------------------------------------------------------------------------
__global__ void gc_gemm_wmma_kernel(const float* __restrict__ x,
                                    const float* __restrict__ W,
                                    float* __restrict__ h) {
    const int lane = threadIdx.x & 31;
    const int wave = threadIdx.x >> 5;      // N tile index 0..7
    const int l    = lane & 15;
    const int half = lane >> 4;             // 0 or 1

    const int row0 = blockIdx.x * 16;       // M tile base (100000 % 16 == 0)
    const int n0   = wave * 16;

    // A: x[row0 + l][k + 2*half .. +1]   (contiguous 8B per lane)
    const float* aPtr = x + (size_t)(row0 + l) * D_IN + half * 2;
    // B: W[k + 2*half + g][n0 + l], g in {0,1} -> stride D_OUT between v0/v1
    const float* bPtr = W + (size_t)(half * 2) * D_OUT + n0 + l;

    v8f c = {};
#pragma unroll 4
    for (int k = 0; k < D_IN; k += 4) {
        v2f a = *(const v2f*)(aPtr + k);
        v2f b;
        b.x = bPtr[(size_t)k * D_OUT];
        b.y = bPtr[(size_t)k * D_OUT + D_OUT];
        // (neg_a, A, neg_b, B, c_mod, C, reuse_a, reuse_b)
        c = __builtin_amdgcn_wmma_f32_16x16x4_f32(
                false, a, false, b, (short)0, c, false, false);
    }

    // Store D: vreg g -> row (row0 + g + 8*half), col n0 + l
    float* hp = h + (size_t)(row0 + half * 8) * D_OUT + n0 + l;
#pragma unroll
    for (int g = 0; g < 8; ++g)
        hp[(size_t)g * D_OUT] = c[g];
}

// ---------------------------------------------------------------------------
// Phase 2: out[r] += v * h[c] over 1.6M COO edges.
// One wave32 per edge, 4 consecutive floats per lane (float4 gather from h,
// 4 f32 atomic adds). h and out both fit in the 192MB L2, so the random
// gather/scatter is L2-resident. 4 edges per wave to amortize launch math.
// ---------------------------------------------------------------------------
#define EDGES_PER_WAVE 4

__global__ void gc_scatter_kernel(const float* __restrict__ h,
                                  const int*   __restrict__ rows,
                                  const int*   __restrict__ cols,
                                  const float* __restrict__ vals,
                                  float* __restrict__ out, int nEdges) {
    const int lane  = threadIdx.x & 31;
    const int waveG = (blockIdx.x * blockDim.x + threadIdx.x) >> 5;
    int e = waveG * EDGES_PER_WAVE;

#pragma unroll
    for (int i = 0; i < EDGES_PER_WAVE; ++i, ++e) {
        if (e >= nEdges) return;
        const int   r = rows[e];      // uniform across wave -> cache broadcast
        const int   c = cols[e];
        const float v = vals[e];

        const float4 m = *(const float4*)(h + (size_t)c * D_OUT + lane * 4);
        float* op = out + (size_t)r * D_OUT + lane * 4;
        atomicAdd(op + 0, v * m.x);
        atomicAdd(op + 1, v * m.y);
        atomicAdd(op + 2, v * m.z);
        atomicAdd(op + 3, v * m.w);
    }
}

// ---------------------------------------------------------------------------
// Phase 3: in-place ReLU.
// ---------------------------------------------------------------------------
__global__ void gc_relu_kernel(float4* __restrict__ out, int n4) {
    int i = blockIdx.x * blockDim.x + threadIdx.x;
    if (i < n4) {
        float4 t = out[i];
        t.x = fmaxf(t.x, 0.f);
        t.y = fmaxf(t.y, 0.f);
        t.z = fmaxf(t.z, 0.f);
        t.w = fmaxf(t.w, 0.f);
        out[i] = t;
    }
}

extern "C" void kernel_launch(void* const* d_in, const int* in_sizes, int n_in,
                              void* d_out, int out_size, void* d_ws, size_t ws_size,
                              hipStream_t stream) {
    const float* x    = (const float*)d_in[0];   // [N_NODES, 256]
    const float* W    = (const float*)d_in[1];   // [256, 128]
    const int*   rows = (const int*)  d_in[2];   // [E]
    const int*   cols = (const int*)  d_in[3];   // [E]
    const float* vals = (const float*)d_in[4];   // [E]
    float*       out  = (float*)d_out;           // [N_NODES, 128]
    float*       h    = (float*)d_ws;            // scratch: [N_NODES, 128]

    const int nEdges = in_sizes[2];
    const int nNodes = in_sizes[0] / D_IN;       // 100000

    const int n4 = out_size / 4;                 // float4 elements
    const int zBlocks = (n4 + 255) / 256;
    gc_zero_kernel<<<zBlocks, 256, 0, stream>>>((float4*)out, n4);

    const int mTiles = nNodes / 16;              // 6250, exact
    gc_gemm_wmma_kernel<<<mTiles, 256, 0, stream>>>(x, W, h);

    // 256 threads = 8 waves = 32 edges per block
    const int edgesPerBlock = 8 * EDGES_PER_WAVE;
    const int sBlocks = (nEdges + edgesPerBlock - 1) / edgesPerBlock;
    gc_scatter_kernel<<<sBlocks, 256, 0, stream>>>(h, rows, cols, vals, out, nEdges);

    gc_relu_kernel<<<zBlocks, 256, 0, stream>>>((float4*)out, n4);
}